// MaskedVideoLoss_68582037782713
// MI455X (gfx1250) — compile-verified
//
#include <hip/hip_runtime.h>
#include <hip/hip_bf16.h>

// ---------------------------------------------------------------------------
// MaskedVideoLoss for MI455X (gfx1250), fp32 end-to-end.
//
// Conv input is spatially constant -> 3x3x3 conv stack computed exactly on the
// 8x8 subgrid {0,1,2,3,60,61,62,63} (boundary radius 3): 64x FLOP reduction.
// GEMMs use V_WMMA_F32_16X16X4_F32 (exact fp32). Conv im2col B-panels are
// staged in LDS via GLOBAL_LOAD_ASYNC_TO_LDS_B32 (ASYNCcnt) and consumed with
// conflict-free ds_load_b64 fragments; weights are pair-interleaved so A
// fragments are single global_load_b64.
// ---------------------------------------------------------------------------

typedef float v2f __attribute__((ext_vector_type(2)));
typedef float v8f __attribute__((ext_vector_type(8)));

#define WMMA4F32(a, b, c) \
  __builtin_amdgcn_wmma_f32_16x16x4_f32(false, (a), false, (b), (short)0, (c), false, false)

__device__ __forceinline__ void async_wait0() {
#if __has_builtin(__builtin_amdgcn_s_wait_asynccnt)
  __builtin_amdgcn_s_wait_asynccnt(0);
#else
  asm volatile("s_wait_asynccnt 0x0" ::: "memory");
#endif
}

// per-lane async global->LDS copy of one dword (GV addressing, saddr=off)
__device__ __forceinline__ void async_copy_b32(float* lds_ptr, const float* gptr) {
  unsigned lo = (unsigned)(uintptr_t)lds_ptr;          // low 32 bits = LDS offset
  unsigned long long ga = (unsigned long long)(uintptr_t)gptr;
  asm volatile("global_load_async_to_lds_b32 %0, %1, off"
               :: "v"(lo), "v"(ga) : "memory");
}

#define NB 4
#define NT 16
#define ND 256
#define KENC 12288           // C*H*W
#define NPOS 1024            // T * 8 * 8 reduced-grid positions

// ---------------- pre-pass: masked obs, pair-interleaved K-major -------------
// store A[m,k] at ((k>>1)*64 + m)*2 + (k&1): lane fragment (k,k+1) is one b64
__global__ __launch_bounds__(256) void k_mask_transpose(
    const float* __restrict__ obs, const unsigned char* __restrict__ mask,
    float* __restrict__ maskedT2) {
  int idx = blockIdx.x * 256 + threadIdx.x;           // over 64*12288
  if (idx >= 64 * KENC) return;
  int m = idx & 63;
  int k = idx >> 6;
  unsigned char mv = mask[m * 4096 + (k & 4095)];
  float v = mv ? 0.0f : obs[(size_t)m * KENC + k];
  maskedT2[(size_t)((k >> 1) * 64 + m) * 2 + (k & 1)] = v;
}

// WencT2: B[k,n] at ((k>>1)*256 + n)*2 + (k&1)
__global__ __launch_bounds__(256) void k_transpose_wenc(
    const float* __restrict__ Wenc, float* __restrict__ WencT2) {
  int idx = blockIdx.x * 256 + threadIdx.x;           // over 12288*256
  if (idx >= KENC * ND) return;
  int n = idx & 255;
  int k = idx >> 8;
  WencT2[(size_t)((k >> 1) * ND + n) * 2 + (k & 1)] = Wenc[(size_t)n * KENC + k];
}

// Wt2: A[o, k=(tap*Cin+d)] at ((k>>1)*CoutP + o)*2 + (k&1), zero-pad o>=Cout
__global__ __launch_bounds__(256) void k_transpose_w(
    const float* __restrict__ W, float* __restrict__ Wt2,
    int Cin, int Cout, int CoutP) {
  int idx = blockIdx.x * 256 + threadIdx.x;           // over 27*Cin*CoutP
  if (idx >= 27 * Cin * CoutP) return;
  int o = idx % CoutP;
  int k = idx / CoutP;
  int tap = k / Cin;
  int d = k % Cin;
  float v = (o < Cout) ? W[((size_t)o * Cin + d) * 27 + tap] : 0.0f;
  Wt2[(size_t)((k >> 1) * CoutP + o) * 2 + (k & 1)] = v;
}

// ---------------- encoder GEMM: feats[m,d], M=64, N=256, K=12288 ------------
__global__ __launch_bounds__(32) void k_enc_gemm(
    const float* __restrict__ maskedT2, const float* __restrict__ WencT2,
    const float* __restrict__ benc, float* __restrict__ feats) {
  int l = threadIdx.x;
  int lm = l & 15;
  int lk2 = (l >> 4) << 1;                            // 0 or 2
  int m = blockIdx.y * 16 + lm;                       // A row
  int n = blockIdx.x * 16 + lm;                       // B col
  v8f acc = {0.f, 0.f, 0.f, 0.f, 0.f, 0.f, 0.f, 0.f};
  for (int kk = 0; kk < KENC; kk += 4) {
    int kb = kk + lk2;                                // even
    v2f a = *(const v2f*)(maskedT2 + (size_t)((kb >> 1) * 64 + m) * 2);
    v2f b = *(const v2f*)(WencT2 + (size_t)((kb >> 1) * ND + n) * 2);
    acc = WMMA4F32(a, b, acc);
  }
#pragma unroll
  for (int v = 0; v < 8; ++v) {
    int row = blockIdx.y * 16 + v + ((l >> 4) << 3);  // M = v + 8*(lane/16)
    int col = blockIdx.x * 16 + (l & 15);             // N = lane%16
    feats[row * ND + col] = acc[v] + benc[col];
  }
}

// ---------------- broadcast feats onto reduced 8x8 grid ---------------------
__global__ __launch_bounds__(256) void k_broadcast(
    const float* __restrict__ feats, float* __restrict__ x0) {
  int idx = blockIdx.x * 256 + threadIdx.x;           // over 4*256*16*64
  if (idx >= NB * ND * NPOS) return;
  int t = (idx >> 6) & 15;
  int d = (idx >> 10) & 255;
  int b = idx >> 18;
  x0[idx] = feats[(b * 16 + t) * ND + d];
}

// ---------------- conv-as-GEMM on reduced grid, LDS-staged B ----------------
// One block per (n-tile, batch); CoutP/16 waves, wave w owns output rows
// [16w,16w+16). Per tap: async-stage X panel (16 x Cin, pad 4) into LDS once,
// then all waves run the WMMA K-loop reading B fragments with ds_load_b64.
__global__ __launch_bounds__(256) void k_conv_gemm(
    const float* __restrict__ Xin, const float* __restrict__ Wt2,
    const float* __restrict__ bias, float* __restrict__ Yout,
    int Cin, int Cout, int CoutP, int doRelu) {
  __shared__ float sX[16 * 260];                      // up to Cin=256, stride Cin+4
  const int stride = Cin + 4;
  int tid = threadIdx.x;
  int l = tid & 31;
  int wv = tid >> 5;                                  // wave id == m-tile
  int lm = l & 15;
  int lk2 = (l >> 4) << 1;
  int b = blockIdx.y;
  int n0 = blockIdx.x * 16;
  int n = n0 + lm;                                    // this lane's B column
  const float* xb = Xin + (size_t)b * Cin * NPOS;
  v8f acc = {0.f, 0.f, 0.f, 0.f, 0.f, 0.f, 0.f, 0.f};

  for (int tap = 0; tap < 27; ++tap) {
    int dt = tap / 9 - 1;
    int r9 = tap % 9;
    int di = r9 / 3 - 1;
    int dj = r9 % 3 - 1;

    __syncthreads();                                  // prior panel fully consumed
    // stage: element e = (d, nn)
    for (int e = tid; e < (Cin << 4); e += blockDim.x) {
      int d = e >> 4;
      int nn = e & 15;
      int np = n0 + nn;
      int t_ = np >> 6, i_ = (np >> 3) & 7, j_ = np & 7;
      int tt = t_ + dt, ii = i_ + di, jj = j_ + dj;
      bool valid = ((unsigned)tt < 16u) & ((unsigned)ii < 8u) & ((unsigned)jj < 8u);
      float* lp = &sX[nn * stride + d];
      if (valid) {
        async_copy_b32(lp, xb + ((size_t)d << 10) + tt * 64 + ii * 8 + jj);
      } else {
        *lp = 0.0f;
      }
    }
    async_wait0();
    __syncthreads();                                  // panel visible to all waves

    const int kbase = tap * Cin;                      // even
    for (int kk = 0; kk < Cin; kk += 4) {
      int d0 = kk + lk2;                              // even
      v2f a = *(const v2f*)(Wt2 + (size_t)(((kbase + d0) >> 1) * CoutP + wv * 16 + lm) * 2);
      v2f bb = *(const v2f*)(&sX[lm * stride + d0]);  // 8B aligned, conflict-free
      acc = WMMA4F32(a, bb, acc);
    }
  }

#pragma unroll
  for (int v = 0; v < 8; ++v) {
    int o = wv * 16 + v + ((l >> 4) << 3);
    if (o < Cout) {
      float rr = acc[v] + bias[o];
      if (doRelu) rr = fmaxf(rr, 0.0f);
      Yout[(((size_t)b * Cout + o) << 10) + n0 + lm] = rr;
    }
  }
}

// ---------------- masked MSE: deterministic two-stage reduction -------------
__global__ __launch_bounds__(256) void k_loss_partial(
    const float* __restrict__ obs, const unsigned char* __restrict__ mask,
    const float* __restrict__ y3, float* __restrict__ partials) {
  __shared__ float ssq[256];
  __shared__ float scnt[256];
  int tid = threadIdx.x;
  int idx = blockIdx.x * 256 + tid;                   // over 4*16*64*64 = 262144
  float sq = 0.0f, cnt = 0.0f;
  unsigned char mv = mask[idx];
  if (mv) {
    cnt = 1.0f;
    int w = idx & 63;
    int h = (idx >> 6) & 63;
    int bt = idx >> 12;
    int t = bt & 15;
    int b = bt >> 4;
    int ri = (h < 3) ? h : ((h > 60) ? h - 56 : 3);   // actual -> reduced row
    int rj = (w < 3) ? w : ((w > 60) ? w - 56 : 3);
    int n = (t << 6) + (ri << 3) + rj;
#pragma unroll
    for (int c = 0; c < 3; ++c) {
      float rv = y3[(((size_t)b * 3 + c) << 10) + n];
      float ov = obs[(((size_t)bt * 3 + c) << 12) + (h << 6) + w];
      float d = rv - ov;
      sq += d * d;
    }
  }
  ssq[tid] = sq;
  scnt[tid] = cnt;
  __syncthreads();
  for (int s = 128; s > 0; s >>= 1) {
    if (tid < s) { ssq[tid] += ssq[tid + s]; scnt[tid] += scnt[tid + s]; }
    __syncthreads();
  }
  if (tid == 0) {
    partials[blockIdx.x * 2 + 0] = ssq[0];
    partials[blockIdx.x * 2 + 1] = scnt[0];
  }
}

__global__ __launch_bounds__(1024) void k_loss_final(
    const float* __restrict__ partials, float* __restrict__ out) {
  __shared__ float ssq[1024];
  __shared__ float scnt[1024];
  int tid = threadIdx.x;
  ssq[tid] = partials[tid * 2 + 0];
  scnt[tid] = partials[tid * 2 + 1];
  __syncthreads();
  for (int s = 512; s > 0; s >>= 1) {
    if (tid < s) { ssq[tid] += ssq[tid + s]; scnt[tid] += scnt[tid + s]; }
    __syncthreads();
  }
  if (tid == 0) {
    float cnt = scnt[0] * 3.0f;                       // count = sum(mask) * C
    out[0] = ssq[0] / fmaxf(cnt, 1.0f);
  }
}

// ---------------------------------------------------------------------------
extern "C" void kernel_launch(void* const* d_in, const int* in_sizes, int n_in,
                              void* d_out, int out_size, void* d_ws, size_t ws_size,
                              hipStream_t stream) {
  (void)in_sizes; (void)n_in; (void)out_size; (void)ws_size;
  const float* obs          = (const float*)d_in[0];
  const unsigned char* mask = (const unsigned char*)d_in[1];   // jnp bool -> 1 byte
  const float* Wenc         = (const float*)d_in[2];
  const float* benc         = (const float*)d_in[3];
  const float* w1 = (const float*)d_in[4];
  const float* b1 = (const float*)d_in[5];
  const float* w2 = (const float*)d_in[6];
  const float* b2 = (const float*)d_in[7];
  const float* w3 = (const float*)d_in[8];
  const float* b3 = (const float*)d_in[9];
  float* out = (float*)d_out;

  // workspace layout (floats)
  float* ws = (float*)d_ws;
  float* maskedT2 = ws;                       // 64*12288      =  786432
  float* WencT2   = maskedT2 + 64 * KENC;     // 12288*256     = 3145728
  float* feats    = WencT2 + KENC * ND;       // 64*256        =   16384
  float* x0       = feats + 64 * ND;          // 4*256*1024    = 1048576
  float* W1t      = x0 + NB * ND * NPOS;      // 27*256*128    =  884736
  float* W2t      = W1t + 27 * 256 * 128;     // 27*128*64     =  221184
  float* W3t      = W2t + 27 * 128 * 64;      // 27*64*16      =   27648
  float* y1       = W3t + 27 * 64 * 16;       // 4*128*1024    =  524288
  float* y2       = y1 + NB * 128 * NPOS;     // 4*64*1024     =  262144
  float* y3       = y2 + NB * 64 * NPOS;      // 4*3*1024      =   12288
  float* partials = y3 + NB * 3 * NPOS;       // 1024*2        =    2048

  // 1) layout pre-passes
  k_mask_transpose<<<(64 * KENC + 255) / 256, 256, 0, stream>>>(obs, mask, maskedT2);
  k_transpose_wenc<<<(KENC * ND + 255) / 256, 256, 0, stream>>>(Wenc, WencT2);
  k_transpose_w<<<(27 * 256 * 128 + 255) / 256, 256, 0, stream>>>(w1, W1t, 256, 128, 128);
  k_transpose_w<<<(27 * 128 * 64 + 255) / 256, 256, 0, stream>>>(w2, W2t, 128, 64, 64);
  k_transpose_w<<<(27 * 64 * 16 + 255) / 256, 256, 0, stream>>>(w3, W3t, 64, 3, 16);

  // 2) encoder GEMM (WMMA f32)
  k_enc_gemm<<<dim3(ND / 16, 64 / 16), 32, 0, stream>>>(maskedT2, WencT2, benc, feats);

  // 3) broadcast to reduced 8x8 grid
  k_broadcast<<<(NB * ND * NPOS + 255) / 256, 256, 0, stream>>>(feats, x0);

  // 4) conv stack on reduced grid (WMMA f32, async-LDS staged im2col)
  k_conv_gemm<<<dim3(NPOS / 16, NB), (128 / 16) * 32, 0, stream>>>(x0, W1t, b1, y1, 256, 128, 128, 1);
  k_conv_gemm<<<dim3(NPOS / 16, NB), ( 64 / 16) * 32, 0, stream>>>(y1, W2t, b2, y2, 128,  64,  64, 1);
  k_conv_gemm<<<dim3(NPOS / 16, NB), ( 16 / 16) * 32, 0, stream>>>(y2, W3t, b3, y3,  64,   3,  16, 0);

  // 5) masked MSE, deterministic two-stage reduction
  k_loss_partial<<<1024, 256, 0, stream>>>(obs, mask, y3, partials);
  k_loss_final<<<1, 1024, 0, stream>>>(partials, out);
}